// LSPE_54975581389179
// MI455X (gfx1250) — compile-verified
//
#include <hip/hip_runtime.h>
#include <hip/hip_bf16.h>

// Problem constants (from reference setup_inputs)
#define BATCH      4096
#define NUM_LABELS 5000
#define NVEC4      (NUM_LABELS / 4)   // 1250 float4/int4 groups per row (exact)

typedef float v2f __attribute__((ext_vector_type(2)));
typedef float v8f __attribute__((ext_vector_type(8)));

#define LOG2E_F 1.44269504088896340736f

__device__ __forceinline__ void acc_one(float x, int tg, float& sn, float& sp) {
    // One transcendental per element: exp(t ? -x : x), then route to the right sum.
    float arg = tg ? -x : x;
    float e   = __builtin_amdgcn_exp2f(arg * LOG2E_F);   // v_exp_f32
    sn += tg ? 0.0f : e;
    sp += tg ? e    : 0.0f;
}

// Stage 1: one 256-thread block per row. Streams b128 loads, accumulates
// s_neg / s_pos per thread, LDS tree reduction, writes log1p(s_neg*s_pos).
__global__ void __launch_bounds__(256)
rowloss_kernel(const float* __restrict__ logits,
               const int*   __restrict__ target,
               float*       __restrict__ row_loss) {
    const int row = blockIdx.x;
    const int tid = threadIdx.x;

    const float4* __restrict__ lrow =
        reinterpret_cast<const float4*>(logits + (size_t)row * NUM_LABELS);
    const int4* __restrict__ trow =
        reinterpret_cast<const int4*>(target + (size_t)row * NUM_LABELS);

    float sn = 0.0f, sp = 0.0f;
    for (int idx = tid; idx < NVEC4; idx += 256) {
        float4 lv = lrow[idx];   // global_load_b128
        int4   tv = trow[idx];   // global_load_b128
        acc_one(lv.x, tv.x, sn, sp);
        acc_one(lv.y, tv.y, sn, sp);
        acc_one(lv.z, tv.z, sn, sp);
        acc_one(lv.w, tv.w, sn, sp);
    }

    __shared__ float red_n[256];
    __shared__ float red_p[256];
    red_n[tid] = sn;
    red_p[tid] = sp;
    __syncthreads();
    #pragma unroll
    for (int s = 128; s > 0; s >>= 1) {
        if (tid < s) {
            red_n[tid] += red_n[tid + s];
            red_p[tid] += red_p[tid + s];
        }
        __syncthreads();
    }

    if (tid == 0) {
        row_loss[row] = log1pf(red_n[0] * red_p[0]);
    }
}

// Stage 2: single wave32 (all lanes active -> EXEC all ones, as WMMA requires).
// Each lane serially sums 4096/32 row losses, then the 32-lane reduction is
// done via v_wmma_f32_16x16x4_f32 with B = ones:
//   A[m][0] = v_m (lanes 0-15, VGPR0), A[m][2] = v_{m+16} (lanes 16-31, VGPR0),
//   A[m][1] = A[m][3] = 0  =>  D[m][n] = v_m + v_{m+16} for every column n.
// Per-lane sum of the 8 D VGPRs gives sum(rowsums 0..7) on lanes 0-15 and
// sum(rowsums 8..15) on lanes 16-31; one xor-16 bpermute + add completes it.
__global__ void __launch_bounds__(32)
final_mean_kernel(const float* __restrict__ row_loss,
                  float*       __restrict__ out) {
    const int lane = threadIdx.x;  // 0..31, all active

    float acc = 0.0f;
    for (int i = lane; i < BATCH; i += 32) {
        acc += row_loss[i];
    }

    v2f a;
    a[0] = acc;   // K=0 (lanes 0-15) / K=2 (lanes 16-31)
    a[1] = 0.0f;  // K=1 / K=3 slots: zero
    v2f b;
    b[0] = 1.0f;  // B = ones(4x16)
    b[1] = 1.0f;
    v8f c = {};

    // 8 args: (neg_a, A, neg_b, B, c_mod, C, reuse_a, reuse_b)
    v8f d = __builtin_amdgcn_wmma_f32_16x16x4_f32(
        false, a, false, b, (short)0, c, false, false);

    float s = d[0] + d[1] + d[2] + d[3] + d[4] + d[5] + d[6] + d[7];

    // Cross-half swap: lane L <-> lane L^16 (wave32 ds_bpermute, byte index)
    int other_bits = __builtin_amdgcn_ds_bpermute((lane ^ 16) << 2,
                                                  __float_as_int(s));
    float total = s + __int_as_float(other_bits);

    if (lane == 0) {
        out[0] = total * (1.0f / (float)BATCH);
    }
}

extern "C" void kernel_launch(void* const* d_in, const int* in_sizes, int n_in,
                              void* d_out, int out_size, void* d_ws, size_t ws_size,
                              hipStream_t stream) {
    const float* logits = (const float*)d_in[0];
    const int*   target = (const int*)d_in[1];
    float*       row_loss = (float*)d_ws;      // BATCH floats of scratch
    float*       out      = (float*)d_out;     // single float (mean loss)

    rowloss_kernel<<<BATCH, 256, 0, stream>>>(logits, target, row_loss);
    final_mean_kernel<<<1, 32, 0, stream>>>(row_loss, out);
}